// NeuralAudioGraph_28724741275811
// MI455X (gfx1250) — compile-verified
//
#include <hip/hip_runtime.h>

typedef __attribute__((ext_vector_type(2))) float v2f;
typedef __attribute__((ext_vector_type(8))) float v8f;

#define NN 256
#define DD 512
#define HH 512
#define EE 128
#define NPAD 516          // 512 + 4 pad: stride%64==4 banks -> conflict-free strided reads
#define LN_EPS 1e-5f

__device__ __forceinline__ float lrelu(float x) { return x > 0.f ? x : 0.2f * x; }

__device__ __forceinline__ v8f wmma_f32(v2f a, v2f b, v8f c) {
    // D = A(16x4 f32) * B(4x16 f32) + C(16x16 f32), wave32
    return __builtin_amdgcn_wmma_f32_16x16x4_f32(false, a, false, b, (short)0, c, false, false);
}

// --- CDNA5 async global->LDS staging (ASYNCcnt-tracked, bypasses VGPRs) ----
__device__ __forceinline__ unsigned lds_off(const void* p) {
    // generic -> LDS(addrspace 3) cast; AS3 pointer value is the byte offset
    // within the workgroup LDS allocation (what VDST of async-to-LDS expects).
    return (unsigned)(uintptr_t)(__attribute__((address_space(3))) const char*)p;
}
__device__ __forceinline__ void async_ld16(const void* gp, unsigned dst) {
    // GV mode: LDS byte offset taken from dst VGPR, global address from the 64-bit gp VGPR pair
    asm volatile("global_load_async_to_lds_b128 %0, %1, off"
                 :: "v"(dst), "v"((unsigned long long)(uintptr_t)gp)
                 : "memory");
}
__device__ __forceinline__ void async_wait0() {
    asm volatile("s_wait_asynccnt 0x0" ::: "memory");
}

// ---------------------------------------------------------------------------
// Generic fp32 WMMA GEMM: C[M x Nc] = A[M x K] @ B[K x Nc] (+ bias).
// Block = 128 threads (4 waves). blockIdx.y -> 16-row tile (A staged in LDS
// via async-to-LDS, shared by the 4 waves); each wave owns one 16-col tile.
// ---------------------------------------------------------------------------
__global__ __launch_bounds__(128) void gemm16(const float* __restrict__ A,
                                              const float* __restrict__ B,
                                              const float* __restrict__ bias,
                                              float* __restrict__ C,
                                              int lda, int K, int ldb, int ldc) {
    __shared__ float sA[16 * NPAD];
    const int t = threadIdx.x;
    const int row0 = blockIdx.y * 16;
    const int kq4 = K >> 2;
    for (int idx = t; idx < 16 * kq4; idx += 128) {
        int r = idx / kq4, kq = (idx - r * kq4) << 2;
        async_ld16(A + (row0 + r) * lda + kq, lds_off(sA + r * NPAD + kq));
    }
    async_wait0();
    __syncthreads();

    const int wave = t >> 5, lane = t & 31;
    const int half = lane >> 4, l15 = lane & 15;
    const int col0 = (blockIdx.x * 4 + wave) * 16;

    v8f acc = {};
    for (int k = 0; k < K; k += 4) {
        const int k0 = k + 2 * half;
        v2f a, b;
        a.x = sA[l15 * NPAD + k0];
        a.y = sA[l15 * NPAD + k0 + 1];
        b.x = B[k0 * ldb + col0 + l15];
        b.y = B[(k0 + 1) * ldb + col0 + l15];
        acc = wmma_f32(a, b, acc);
    }
    const int n = col0 + l15;
    const float bv = bias ? bias[n] : 0.f;
#pragma unroll
    for (int r = 0; r < 8; r++)
        C[(row0 + r + 8 * half) * ldc + n] = acc[r] + bv;
}

// ---------------------------------------------------------------------------
// Pair kernel: one block per 16x16 (i,j) tile (grid 16x16, 256 threads/8 waves)
// Phase 1: edge predictor  -> active[i,j] (d_out), adj[j,i] (ws mask)
// Phase 2: edge feature net -> LN stats, then [256 pair x 512] @ ef_w2[512x128]
//          via fp32 WMMA with LN/LeakyReLU A-fragments built on the fly.
// ---------------------------------------------------------------------------
__global__ __launch_bounds__(256) void pair_kernel(
    const float* __restrict__ epA, const float* __restrict__ epB,
    const float* __restrict__ efA, const float* __restrict__ efB,
    const float* __restrict__ ep_g, const float* __restrict__ ep_be,
    const float* __restrict__ ep_w2, const float* __restrict__ ep_b2,
    const float* __restrict__ ef_g, const float* __restrict__ ef_be,
    const float* __restrict__ ef_w2, const float* __restrict__ ef_b2,
    float* __restrict__ out_active, float* __restrict__ adj,
    float* __restrict__ edge_out) {
    extern __shared__ float smem[];
    float* sA = smem;                 // 16*NPAD : A rows (i tile)
    float* sB = sA + 16 * NPAD;       // 16*NPAD : B rows (j tile, bias folded)
    float* sG = sB + 16 * NPAD;       // 512 : LN gamma
    float* sBe = sG + 512;            // 512 : LN beta
    float* sW2 = sBe + 512;           // 512 : ep_w2
    float* sMean = sW2 + 512;         // 256 : per-pair LN mean  (ef)
    float* sRstd = sMean + 256;       // 256 : per-pair LN rstd  (ef)
    float* sB2 = sRstd + 256;         // 128 : ef_b2

    const int t = threadIdx.x;
    const int i0 = blockIdx.y * 16, j0 = blockIdx.x * 16;
    const int ii = t >> 4, jj = t & 15;   // one thread per pair in the tile

    // pull the ef second-layer weights (256KB, shared by all blocks) toward L2
    __builtin_prefetch(ef_w2 + t * 256, 0, 0);

    // ---------------- phase 1: edge predictor ----------------
    for (int idx = t; idx < 16 * 128; idx += 256) {
        int r = idx >> 7, kq = (idx & 127) << 2;
        async_ld16(epA + (i0 + r) * HH + kq, lds_off(sA + r * NPAD + kq));
        async_ld16(epB + (j0 + r) * HH + kq, lds_off(sB + r * NPAD + kq));
    }
    for (int k = t; k < 512; k += 256) {
        sG[k] = ep_g[k]; sBe[k] = ep_be[k]; sW2[k] = ep_w2[k];
    }
    async_wait0();
    __syncthreads();
    {
        float sum = 0.f, sq = 0.f;
        for (int k = 0; k < HH; k++) {
            float v = sA[ii * NPAD + k] + sB[jj * NPAD + k];
            sum += v; sq += v * v;
        }
        const float mean = sum * (1.f / HH);
        const float var = sq * (1.f / HH) - mean * mean;
        const float rstd = rsqrtf(var + LN_EPS);
        float dot = 0.f;
        for (int k = 0; k < HH; k++) {
            float v = sA[ii * NPAD + k] + sB[jj * NPAD + k];
            float h = lrelu((v - mean) * rstd * sG[k] + sBe[k]);
            dot += h * sW2[k];
        }
        const float logit = dot + ep_b2[0];          // sigmoid(l)>0.5 <=> l>0
        const int i = i0 + ii, j = j0 + jj;
        const bool act = (logit > 0.f) && (i != j);
        out_active[i * NN + j] = act ? 1.f : 0.f;
        adj[j * NN + i] = (act || i == j) ? 1.f : 0.f;   // adj = active.T | eye
    }
    __syncthreads();

    // ---------------- phase 2: edge feature net ----------------
    for (int idx = t; idx < 16 * 128; idx += 256) {
        int r = idx >> 7, kq = (idx & 127) << 2;
        async_ld16(efA + (i0 + r) * HH + kq, lds_off(sA + r * NPAD + kq));
        async_ld16(efB + (j0 + r) * HH + kq, lds_off(sB + r * NPAD + kq));
    }
    for (int k = t; k < 512; k += 256) { sG[k] = ef_g[k]; sBe[k] = ef_be[k]; }
    if (t < EE) sB2[t] = ef_b2[t];
    async_wait0();
    __syncthreads();
    {
        float sum = 0.f, sq = 0.f;
        for (int k = 0; k < HH; k++) {
            float v = sA[ii * NPAD + k] + sB[jj * NPAD + k];
            sum += v; sq += v * v;
        }
        const float mean = sum * (1.f / HH);
        const float var = sq * (1.f / HH) - mean * mean;
        sMean[t] = mean;
        sRstd[t] = rsqrtf(var + LN_EPS);
    }
    __syncthreads();

    // WMMA: wave w owns pairs p = [32w, 32w+32) as 2 M-tiles; full E=128 (8 N-tiles)
    const int wave = t >> 5, lane = t & 31;
    const int half = lane >> 4, l15 = lane & 15;
    v8f acc[2][8];
#pragma unroll
    for (int mt = 0; mt < 2; mt++)
#pragma unroll
        for (int nt = 0; nt < 8; nt++) acc[mt][nt] = {};

    for (int k = 0; k < HH; k += 4) {
        const int k0 = k + 2 * half;
        v2f bf[8];
#pragma unroll
        for (int nt = 0; nt < 8; nt++) {   // ef_w2 rows k0,k0+1 (L2-resident)
            bf[nt].x = ef_w2[k0 * EE + nt * 16 + l15];
            bf[nt].y = ef_w2[(k0 + 1) * EE + nt * 16 + l15];
        }
        const float g0 = sG[k0], g1 = sG[k0 + 1];
        const float be0 = sBe[k0], be1 = sBe[k0 + 1];
        const float vb0 = sB[l15 * NPAD + k0], vb1 = sB[l15 * NPAD + k0 + 1];
#pragma unroll
        for (int mt = 0; mt < 2; mt++) {
            const int m = wave * 2 + mt;                 // pair = m*16 + l15 -> ii=m, jj=l15
            const float mn = sMean[m * 16 + l15], rs = sRstd[m * 16 + l15];
            v2f af;
            af.x = lrelu((sA[m * NPAD + k0] + vb0 - mn) * rs * g0 + be0);
            af.y = lrelu((sA[m * NPAD + k0 + 1] + vb1 - mn) * rs * g1 + be1);
#pragma unroll
            for (int nt = 0; nt < 8; nt++)
                acc[mt][nt] = wmma_f32(af, bf[nt], acc[mt][nt]);
        }
    }

#pragma unroll
    for (int mt = 0; mt < 2; mt++) {
        const int m = wave * 2 + mt;
        const int i = i0 + m;
#pragma unroll
        for (int nt = 0; nt < 8; nt++) {
            const int e = nt * 16 + l15;
            const float b2 = sB2[e];
#pragma unroll
            for (int r = 0; r < 8; r++) {
                const int j = j0 + r + 8 * half;         // C: M = r + 8*half
                edge_out[(i * NN + j) * EE + e] = acc[mt][nt][r] + b2;
            }
        }
    }
}

// ---------------------------------------------------------------------------
// GAT attention scores: one wave per (n,h): a_s/a_d dot products over C=512.
// ---------------------------------------------------------------------------
__global__ __launch_bounds__(128) void gat_scores(const float* __restrict__ s,
                                                  const float* __restrict__ wsrc,
                                                  const float* __restrict__ wdst,
                                                  float* __restrict__ a_s,
                                                  float* __restrict__ a_d) {
    const int t = threadIdx.x;
    const int g = blockIdx.x * 4 + (t >> 5);   // 0..1023
    const int lane = t & 31;
    const int n = g >> 2, h = g & 3;
    const float* sp = s + n * (4 * HH) + h * HH;
    const float* ws = wsrc + h * HH;
    const float* wd = wdst + h * HH;
    float accs = 0.f, accd = 0.f;
    for (int k = lane; k < HH; k += 32) {
        float v = sp[k];
        accs += v * ws[k];
        accd += v * wd[k];
    }
    for (int off = 16; off > 0; off >>= 1) {
        accs += __shfl_xor(accs, off, 32);
        accd += __shfl_xor(accd, off, 32);
    }
    if (lane == 0) { a_s[n * 4 + h] = accs; a_d[n * 4 + h] = accd; }
}

// ---------------------------------------------------------------------------
// Masked softmax over sources: one wave per (dst i, head h), 8 j's per lane.
// ---------------------------------------------------------------------------
__global__ __launch_bounds__(128) void gat_softmax(const float* __restrict__ a_s,
                                                   const float* __restrict__ a_d,
                                                   const float* __restrict__ adj,
                                                   float* __restrict__ alpha) {
    const int t = threadIdx.x;
    const int g = blockIdx.x * 4 + (t >> 5);
    const int lane = t & 31;
    const int i = g >> 2, h = g & 3;
    const float ad = a_d[i * 4 + h];
    float lg[8], mk[8], e[8];
    float vmax = -3.0e38f;
#pragma unroll
    for (int tt = 0; tt < 8; tt++) {
        const int j = lane + 32 * tt;
        const float l = lrelu(ad + a_s[j * 4 + h]);
        const float m = adj[i * NN + j];
        lg[tt] = l; mk[tt] = m;
        vmax = (m > 0.f && l > vmax) ? l : vmax;   // diag always active
    }
    for (int off = 16; off > 0; off >>= 1) vmax = fmaxf(vmax, __shfl_xor(vmax, off, 32));
    float sum = 0.f;
#pragma unroll
    for (int tt = 0; tt < 8; tt++) {
        const float v = (mk[tt] > 0.f) ? __expf(lg[tt] - vmax) : 0.f;
        e[tt] = v; sum += v;
    }
    for (int off = 16; off > 0; off >>= 1) sum += __shfl_xor(sum, off, 32);
    const float inv = 1.f / sum;
#pragma unroll
    for (int tt = 0; tt < 8; tt++)
        alpha[h * NN * NN + i * NN + lane + 32 * tt] = e[tt] * inv;
}

// ---------------------------------------------------------------------------
// x_new[i,c] = lrelu( 0.25 * sum_h (alpha_h @ s_h)[i,c] + b[c] )  via WMMA.
// ---------------------------------------------------------------------------
__global__ __launch_bounds__(128) void gat_agg(const float* __restrict__ alpha,
                                               const float* __restrict__ s,
                                               const float* __restrict__ b,
                                               float* __restrict__ xout) {
    const int t = threadIdx.x;
    const int wave = t >> 5, lane = t & 31;
    const int half = lane >> 4, l15 = lane & 15;
    const int row0 = blockIdx.y * 16;
    const int col0 = (blockIdx.x * 4 + wave) * 16;
    v8f acc = {};
    for (int h = 0; h < 4; h++) {
        const float* Ah = alpha + h * NN * NN;    // [256 dst x 256 src]
        const float* Bh = s + h * HH;             // rows stride 2048
        for (int k = 0; k < NN; k += 4) {
            const int k0 = k + 2 * half;
            v2f a, bf;
            a.x = Ah[(row0 + l15) * NN + k0];
            a.y = Ah[(row0 + l15) * NN + k0 + 1];
            bf.x = Bh[k0 * (4 * HH) + col0 + l15];
            bf.y = Bh[(k0 + 1) * (4 * HH) + col0 + l15];
            acc = wmma_f32(a, bf, acc);
        }
    }
    const int n = col0 + l15;
    const float bv = b[n];
#pragma unroll
    for (int r = 0; r < 8; r++)
        xout[(row0 + r + 8 * half) * HH + n] = lrelu(acc[r] * 0.25f + bv);
}

// ---------------------------------------------------------------------------
// In-place LayerNorm(g,beta) + LeakyReLU over rows of [256 x 512].
// ---------------------------------------------------------------------------
__global__ __launch_bounds__(256) void ln_lrelu_kernel(float* __restrict__ x,
                                                       const float* __restrict__ g,
                                                       const float* __restrict__ be) {
    __shared__ float rs[8], rq[8], stat[2];
    const int row = blockIdx.x, t = threadIdx.x;
    const float v0 = x[row * HH + t], v1 = x[row * HH + t + 256];
    float s = v0 + v1, q = v0 * v0 + v1 * v1;
    for (int off = 16; off > 0; off >>= 1) {
        s += __shfl_xor(s, off, 32);
        q += __shfl_xor(q, off, 32);
    }
    const int wave = t >> 5, lane = t & 31;
    if (lane == 0) { rs[wave] = s; rq[wave] = q; }
    __syncthreads();
    if (t == 0) {
        float S = 0.f, Q = 0.f;
        for (int w = 0; w < 8; w++) { S += rs[w]; Q += rq[w]; }
        const float mean = S * (1.f / HH);
        stat[0] = mean;
        stat[1] = rsqrtf(Q * (1.f / HH) - mean * mean + LN_EPS);
    }
    __syncthreads();
    const float mean = stat[0], rstd = stat[1];
    x[row * HH + t] = lrelu((v0 - mean) * rstd * g[t] + be[t]);
    x[row * HH + t + 256] = lrelu((v1 - mean) * rstd * g[t + 256] + be[t + 256]);
}

// ---------------------------------------------------------------------------
extern "C" void kernel_launch(void* const* d_in, const int* in_sizes, int n_in,
                              void* d_out, int out_size, void* d_ws, size_t ws_size,
                              hipStream_t stream) {
    const float* emb   = (const float*)d_in[0];
    const float* ep_w1 = (const float*)d_in[1];
    const float* ep_b1 = (const float*)d_in[2];
    const float* ep_g  = (const float*)d_in[3];
    const float* ep_be = (const float*)d_in[4];
    const float* ep_w2 = (const float*)d_in[5];
    const float* ep_b2 = (const float*)d_in[6];
    const float* ef_w1 = (const float*)d_in[7];
    const float* ef_b1 = (const float*)d_in[8];
    const float* ef_g  = (const float*)d_in[9];
    const float* ef_be = (const float*)d_in[10];
    const float* ef_w2 = (const float*)d_in[11];
    const float* ef_b2 = (const float*)d_in[12];
    const float* gat_W = (const float*)d_in[13];
    const float* gat_as= (const float*)d_in[14];
    const float* gat_ad= (const float*)d_in[15];
    const float* gat_b = (const float*)d_in[16];
    const float* op_w1 = (const float*)d_in[17];
    const float* op_b1 = (const float*)d_in[18];
    const float* op_g  = (const float*)d_in[19];
    const float* op_be = (const float*)d_in[20];
    const float* op_w2 = (const float*)d_in[21];
    const float* op_b2 = (const float*)d_in[22];

    float* out    = (float*)d_out;        // [256,512]
    float* active = out + NN * DD;        // [256,256]
    float* edge   = active + NN * NN;     // [256,256,128]

    float* w = (float*)d_ws;
    float* epA  = w; w += NN * HH;
    float* epB  = w; w += NN * HH;
    float* efA  = w; w += NN * HH;
    float* efB  = w; w += NN * HH;
    float* adj  = w; w += NN * NN;
    float* sbuf = w; w += NN * 4 * HH;
    float* a_s  = w; w += NN * 4;
    float* a_d  = w; w += NN * 4;
    float* alph = w; w += 4 * NN * NN;
    float* x1   = w; w += NN * HH;
    float* x2   = w; w += NN * HH;
    float* hb   = w; w += NN * HH;

    const dim3 b128(128), b256(256);
    const dim3 g512(8, 16), g2048(32, 16);

    // factorized first layers of both pair MLPs (bias folded into B half)
    gemm16<<<g512, b128, 0, stream>>>(emb, ep_w1,           nullptr, epA, DD, DD, HH, HH);
    gemm16<<<g512, b128, 0, stream>>>(emb, ep_w1 + DD * HH, ep_b1,   epB, DD, DD, HH, HH);
    gemm16<<<g512, b128, 0, stream>>>(emb, ef_w1,           nullptr, efA, DD, DD, HH, HH);
    gemm16<<<g512, b128, 0, stream>>>(emb, ef_w1 + DD * HH, ef_b1,   efB, DD, DD, HH, HH);

    const size_t smem = (size_t)(2 * 16 * NPAD + 3 * 512 + 2 * 256 + 128) * sizeof(float);
    pair_kernel<<<dim3(16, 16), b256, smem, stream>>>(epA, epB, efA, efB,
        ep_g, ep_be, ep_w2, ep_b2, ef_g, ef_be, ef_w2, ef_b2, active, adj, edge);

    const float* xin = emb;
    for (int l = 0; l < 3; l++) {
        float* xout = (l == 1) ? x2 : x1;
        gemm16<<<g2048, b128, 0, stream>>>(xin, gat_W + (size_t)l * HH * 4 * HH,
                                           nullptr, sbuf, HH, HH, 4 * HH, 4 * HH);
        gat_scores<<<256, b128, 0, stream>>>(sbuf, gat_as + l * 4 * HH, gat_ad + l * 4 * HH, a_s, a_d);
        gat_softmax<<<256, b128, 0, stream>>>(a_s, a_d, adj, alph);
        gat_agg<<<dim3(8, 16), b128, 0, stream>>>(alph, sbuf, gat_b + l * HH, xout);
        xin = xout;
    }

    gemm16<<<g512, b128, 0, stream>>>(xin, op_w1, op_b1, hb, HH, HH, HH, HH);
    ln_lrelu_kernel<<<256, b256, 0, stream>>>(hb, op_g, op_be);
    gemm16<<<g512, b128, 0, stream>>>(hb, op_w2, op_b2, out, HH, HH, DD, DD);
}